// VRNN_32160715112779
// MI455X (gfx1250) — compile-verified
//
#include <hip/hip_runtime.h>
#include <cmath>

typedef float v2f __attribute__((ext_vector_type(2)));
typedef float v8f __attribute__((ext_vector_type(8)));
typedef unsigned tdm_g0_t __attribute__((ext_vector_type(4)));
typedef int tdm_g1_t __attribute__((ext_vector_type(8)));
typedef int tdm_g4_t __attribute__((ext_vector_type(4)));

#define V_SZ   32000
#define E_DIM  256
#define H_SZ   1024
#define BATCH  8
#define SEQ    512
#define M_ROWS (BATCH * SEQ)   // 4096

// ---------------- recurrence persistent-kernel config ----------------
#define NBLK 64       // one WGP each (256 KB LDS/block); MI455X has >= 64 WGPs
#define CPB  16       // H columns owned per block (64 * 16 = 1024)
#define WROW 1028     // padded LDS row stride (floats): 1028 % 64 = 4 -> bank-skewed

// ---------------------------------------------------------------------
// LDS byte offset of a __shared__ object (generic -> addrspace(3) -> int).
// Used only as a scalar value (never in aggregate initializers, which the
// linker cannot relocate).
// ---------------------------------------------------------------------
__device__ __forceinline__ unsigned lds_off(const void* p) {
  typedef __attribute__((address_space(3))) const void lds_cv;
  return (unsigned)(unsigned long long)(lds_cv*)p;
}

// ---------------------------------------------------------------------
// Tensor Data Mover: 2-D tile load (global -> LDS), D# per ISA ch.8.
// tile rows: tile_d0 contiguous 4-byte elements, tile_d1 rows, row stride
// stride0 elements. Optional LDS row padding: after every 2^(pad_ivl+1)
// DWORDs insert (pad_amt+1) DWORDs. Tiles are always fully in-bounds, so
// tensor dims == tile dims. Tracked by TENSORcnt.
// ---------------------------------------------------------------------
__device__ __forceinline__ void tdm_load_2d(unsigned ldsOff, const void* gptr,
                                            unsigned tile_d0, unsigned tile_d1,
                                            unsigned long long stride0,
                                            unsigned pad_en, unsigned pad_ivl,
                                            unsigned pad_amt) {
  const unsigned long long ga = (unsigned long long)gptr;
  tdm_g0_t g0;
  g0[0] = 1u;                                   // count=1, user descriptor
  g0[1] = ldsOff;                               // lds_addr
  g0[2] = (unsigned)ga;                         // global_addr[31:0]
  g0[3] = (unsigned)((ga >> 32) & 0x01FFFFFFu)  // global_addr[56:32]
          | (2u << 30);                         // type = 2 ("image")
  tdm_g1_t g1;
  g1[0] = (int)((2u << 16) |                    // data_size = 4 bytes
                (pad_en << 20) | (pad_ivl << 22) | (pad_amt << 25));
  g1[1] = (int)((tile_d0 & 0xFFFFu) << 16);     // tensor_dim0[15:0]
  g1[2] = (int)((tile_d0 >> 16) | ((tile_d1 & 0xFFFFu) << 16)); // d0 hi | d1 lo
  g1[3] = (int)((tile_d1 >> 16) | (tile_d0 << 16));             // d1 hi | tile_dim0
  g1[4] = (int)(tile_d1 & 0xFFFFu);             // tile_dim1 (tile_dim2 = 0)
  g1[5] = (int)(unsigned)stride0;               // tensor_dim0_stride[31:0]
  g1[6] = (int)((stride0 >> 32) & 0xFFFFu);     // stride0[47:32]
  g1[7] = 0;
  const tdm_g4_t z4 = {0, 0, 0, 0};
#if defined(__clang_major__) && (__clang_major__ >= 23)
  const tdm_g1_t z8 = {0, 0, 0, 0, 0, 0, 0, 0};
  __builtin_amdgcn_tensor_load_to_lds(g0, g1, z4, z4, z8, 0);
#else
  __builtin_amdgcn_tensor_load_to_lds(g0, g1, z4, z4, 0);
#endif
}

// ---------------------------------------------------------------------
// Grid-wide sense barrier (generation counter in global ws memory).
// ---------------------------------------------------------------------
__device__ __forceinline__ void grid_sync(unsigned* bar) {
  __syncthreads();
  if (threadIdx.x == 0) {
    __threadfence();
    unsigned g = __hip_atomic_load(&bar[1], __ATOMIC_ACQUIRE,
                                   __HIP_MEMORY_SCOPE_AGENT);
    unsigned arrived = __hip_atomic_fetch_add(&bar[0], 1u, __ATOMIC_ACQ_REL,
                                              __HIP_MEMORY_SCOPE_AGENT);
    if (arrived == NBLK - 1u) {
      __hip_atomic_store(&bar[0], 0u, __ATOMIC_RELAXED,
                         __HIP_MEMORY_SCOPE_AGENT);
      __hip_atomic_fetch_add(&bar[1], 1u, __ATOMIC_RELEASE,
                             __HIP_MEMORY_SCOPE_AGENT);
    } else {
      while (__hip_atomic_load(&bar[1], __ATOMIC_ACQUIRE,
                               __HIP_MEMORY_SCOPE_AGENT) == g)
        __builtin_amdgcn_s_sleep(2);
    }
  }
  __syncthreads();
}

// ---------------------------------------------------------------------
// Kernel 1: X0 = emb[x] @ W_ih0 + (b_ih0 + b_hh0)      [4096 x 1024]
// (embedding gather fused into A staging; WMMA f32 16x16x4)
// ---------------------------------------------------------------------
__global__ __launch_bounds__(256) void k_x0(
    const int* __restrict__ x, const float* __restrict__ emb,
    const float* __restrict__ W_ih0, const float* __restrict__ b_ih0,
    const float* __restrict__ b_hh0, float* __restrict__ X0) {
  __shared__ float As[64][18];
  __shared__ float Bs[16][132];

  const int tid  = threadIdx.x;
  const int lane = tid & 31;
  const int w    = tid >> 5;
  const int wm   = w >> 2, wn = w & 3;
  const int mbase = blockIdx.y * 64;
  const int nbase = blockIdx.x * 128;
  const int lm = lane & 15;
  const int kh = lane >> 4;

  v8f acc[2][2] = {};

  for (int kt = 0; kt < E_DIM; kt += 16) {
    __syncthreads();
    for (int i = tid; i < 64 * 16; i += 256) {
      int r = i >> 4, k = i & 15;
      int tok = x[mbase + r];
      As[r][k] = emb[tok * E_DIM + kt + k];
    }
    for (int i = tid; i < 16 * 128; i += 256) {
      int k = i >> 7, n = i & 127;
      Bs[k][n] = W_ih0[(kt + k) * H_SZ + nbase + n];
    }
    __syncthreads();

#pragma unroll
    for (int kc = 0; kc < 4; ++kc) {
      const int k0 = kc * 4 + kh * 2;
      v2f a[2], b[2];
#pragma unroll
      for (int mt = 0; mt < 2; ++mt) {
        const float* ap = &As[wm * 32 + mt * 16 + lm][k0];
        a[mt].x = ap[0];
        a[mt].y = ap[1];
      }
#pragma unroll
      for (int nt = 0; nt < 2; ++nt) {
        const int n = wn * 32 + nt * 16 + lm;
        b[nt].x = Bs[k0][n];
        b[nt].y = Bs[k0 + 1][n];
      }
#pragma unroll
      for (int mt = 0; mt < 2; ++mt)
#pragma unroll
        for (int nt = 0; nt < 2; ++nt)
          acc[mt][nt] = __builtin_amdgcn_wmma_f32_16x16x4_f32(
              false, a[mt], false, b[nt], (short)0, acc[mt][nt], false, false);
    }
  }

#pragma unroll
  for (int mt = 0; mt < 2; ++mt)
#pragma unroll
    for (int nt = 0; nt < 2; ++nt) {
      const int col = nbase + wn * 32 + nt * 16 + lm;
      const float bias = b_ih0[col] + b_hh0[col];
#pragma unroll
      for (int v = 0; v < 8; ++v) {
        const int row = mbase + wm * 32 + mt * 16 + kh * 8 + v;
        X0[(size_t)row * H_SZ + col] = acc[mt][nt][v] + bias;
      }
    }
}

// ---------------------------------------------------------------------
// Kernel 2: persistent recurrence (64 blocks x 256 thr, weights in LDS,
// one grid sync per timestep, h1(t) & h0(t+1) computed concurrently).
// ---------------------------------------------------------------------
__global__ __launch_bounds__(256) void k_recur(
    const float* __restrict__ X0, const float* __restrict__ W_hh0,
    const float* __restrict__ W_ih1, const float* __restrict__ b_ih1,
    const float* __restrict__ W_hh1, const float* __restrict__ b_hh1,
    float* __restrict__ outs, float* __restrict__ hbuf, unsigned* bar) {
  extern __shared__ float smem[];
  float* wh0 = smem;                 // [CPB][WROW]
  float* wi1 = wh0 + CPB * WROW;
  float* wh1 = wi1 + CPB * WROW;
  float* h0s = wh1 + CPB * WROW;     // [BATCH][WROW]
  float* h1s = h0s + BATCH * WROW;

  const int tid = threadIdx.x;
  const int colbase = blockIdx.x * CPB;

  for (int i = tid; i < CPB * H_SZ; i += 256) {
    int c = i & (CPB - 1), k = i >> 4;
    wh0[c * WROW + k] = W_hh0[k * H_SZ + colbase + c];
    wi1[c * WROW + k] = W_ih1[k * H_SZ + colbase + c];
    wh1[c * WROW + k] = W_hh1[k * H_SZ + colbase + c];
  }

  const int pair = tid >> 1;
  const int b = pair >> 4;
  const int c = pair & 15;
  const int kh = tid & 1;
  const int col = colbase + c;
  const float bias1 = b_ih1[col] + b_hh1[col];

  float* h0g = hbuf;
  float* h1g = hbuf + 2 * BATCH * H_SZ;

  if (kh == 0) {
    h0g[b * H_SZ + col] = tanhf(X0[(size_t)(b * SEQ) * H_SZ + col]);
    h1g[b * H_SZ + col] = 0.f;
  }
  grid_sync(bar);

  for (int t = 0; t < SEQ; ++t) {
    const int pr = t & 1, pw = pr ^ 1;

    for (int i = tid; i < BATCH * H_SZ; i += 256) {
      int bb = i >> 10, k = i & 1023;
      h0s[bb * WROW + k] = h0g[pr * BATCH * H_SZ + i];
      h1s[bb * WROW + k] = h1g[pr * BATCH * H_SZ + i];
    }
    __syncthreads();

    const float4* hp0 = (const float4*)&h0s[b * WROW + kh * 512];
    const float4* hp1 = (const float4*)&h1s[b * WROW + kh * 512];
    const float4* w0p = (const float4*)&wh0[c * WROW + kh * 512];
    const float4* w1p = (const float4*)&wi1[c * WROW + kh * 512];
    const float4* w2p = (const float4*)&wh1[c * WROW + kh * 512];
    float d0 = 0.f, d1 = 0.f, d2 = 0.f;
#pragma unroll 4
    for (int kk = 0; kk < 128; ++kk) {
      float4 h0v = hp0[kk], h1v = hp1[kk];
      float4 a0 = w0p[kk], a1 = w1p[kk], a2 = w2p[kk];
      d0 += h0v.x * a0.x + h0v.y * a0.y + h0v.z * a0.z + h0v.w * a0.w;
      d1 += h0v.x * a1.x + h0v.y * a1.y + h0v.z * a1.z + h0v.w * a1.w;
      d2 += h1v.x * a2.x + h1v.y * a2.y + h1v.z * a2.z + h1v.w * a2.w;
    }
    d0 += __shfl_xor(d0, 1, 32);
    d1 += __shfl_xor(d1, 1, 32);
    d2 += __shfl_xor(d2, 1, 32);

    if (kh == 0) {
      const float h1n = tanhf(d1 + bias1 + d2);
      outs[(size_t)(b * SEQ + t) * H_SZ + col] = h1n;
      h1g[pw * BATCH * H_SZ + b * H_SZ + col] = h1n;
      if (t + 1 < SEQ) {
        const float h0n =
            tanhf(X0[(size_t)(b * SEQ + t + 1) * H_SZ + col] + d0);
        h0g[pw * BATCH * H_SZ + b * H_SZ + col] = h0n;
      }
    }
    __syncthreads();
    grid_sync(bar);
  }
}

// ---------------------------------------------------------------------
// Kernel 3: logits = outs @ W_out + b_out   [4096 x 32000], K=1024.
// Store-bound. Tiles DMA'd by the Tensor Data Mover straight into LDS
// (hardware row padding -> bank-skewed strides 18 / 132), double-buffered
// so TDM(tile k+1) overlaps WMMA(tile k). Wave 0 issues; all waves gate
// on s_wait_tensorcnt + s_barrier. All LDS addresses kept scalar.
// ---------------------------------------------------------------------
#define ASZ (64 * 18)
#define BSZ (16 * 132)

__global__ __launch_bounds__(256) void k_logits(
    const float* __restrict__ A, const float* __restrict__ W_out,
    const float* __restrict__ b_out, float* __restrict__ out) {
  __shared__ float smem[2 * ASZ + 2 * BSZ];

  const int tid  = threadIdx.x;
  const int lane = tid & 31;
  const int w    = tid >> 5;
  const int wm   = w >> 2, wn = w & 3;
  const int mbase = blockIdx.y * 64;
  const int nbase = blockIdx.x * 128;
  const int lm = lane & 15;
  const int kh = lane >> 4;

  const unsigned ldsBase = lds_off(smem);   // scalar only; never in aggregates

  v8f acc[2][2] = {};
  const int NT = H_SZ / 16;   // 64 K-tiles

  if (tid < 32) {  // wave 0: prime the pipeline with tile 0
    tdm_load_2d(ldsBase, &A[(size_t)mbase * H_SZ], 16, 64, H_SZ, 1, 3, 1);
    tdm_load_2d(ldsBase + 2u * ASZ * 4u, &W_out[(size_t)0 * V_SZ + nbase],
                128, 16, V_SZ, 1, 6, 3);
  }

  for (int it = 0; it < NT; ++it) {
    const int p = it & 1;
    const int kt = it * 16;
    if (tid < 32) __builtin_amdgcn_s_wait_tensorcnt(0);
    __syncthreads();                      // tile `it` resident in LDS
    if (it + 1 < NT && tid < 32) {        // overlap DMA of next tile
      const unsigned q = (unsigned)(p ^ 1);
      tdm_load_2d(ldsBase + q * (ASZ * 4u),
                  &A[(size_t)mbase * H_SZ + kt + 16], 16, 64, H_SZ, 1, 3, 1);
      tdm_load_2d(ldsBase + 2u * ASZ * 4u + q * (BSZ * 4u),
                  &W_out[(size_t)(kt + 16) * V_SZ + nbase], 128, 16, V_SZ, 1,
                  6, 3);
    }
    if (kt + 32 < H_SZ)                   // warm L2 two tiles ahead
      __builtin_prefetch(&W_out[(size_t)(kt + 32 + (tid >> 4)) * V_SZ + nbase +
                                ((tid & 15) << 3)], 0, 1);

    const float* Ac = smem + p * ASZ;
    const float* Bc = smem + 2 * ASZ + p * BSZ;
#pragma unroll
    for (int kc = 0; kc < 4; ++kc) {
      const int k0 = kc * 4 + kh * 2;
      v2f a[2], b[2];
#pragma unroll
      for (int mt = 0; mt < 2; ++mt) {
        const float* ap = &Ac[(wm * 32 + mt * 16 + lm) * 18 + k0];
        a[mt].x = ap[0];
        a[mt].y = ap[1];
      }
#pragma unroll
      for (int nt = 0; nt < 2; ++nt) {
        const int n = wn * 32 + nt * 16 + lm;
        b[nt].x = Bc[k0 * 132 + n];
        b[nt].y = Bc[(k0 + 1) * 132 + n];
      }
#pragma unroll
      for (int mt = 0; mt < 2; ++mt)
#pragma unroll
        for (int nt = 0; nt < 2; ++nt)
          acc[mt][nt] = __builtin_amdgcn_wmma_f32_16x16x4_f32(
              false, a[mt], false, b[nt], (short)0, acc[mt][nt], false, false);
    }
    __syncthreads();                      // reads done before buffer reuse
  }

#pragma unroll
  for (int mt = 0; mt < 2; ++mt)
#pragma unroll
    for (int nt = 0; nt < 2; ++nt) {
      const int col = nbase + wn * 32 + nt * 16 + lm;
      const float bias = b_out[col];
#pragma unroll
      for (int v = 0; v < 8; ++v) {
        const int row = mbase + wm * 32 + mt * 16 + kh * 8 + v;
        out[(size_t)row * V_SZ + col] = acc[mt][nt][v] + bias;
      }
    }
}

// ---------------------------------------------------------------------
// Kernel 4: emit final hidden states (h0 in parity 1, h1 in parity 0)
// ---------------------------------------------------------------------
__global__ void k_tail(const float* __restrict__ hbuf,
                       float* __restrict__ out_h0,
                       float* __restrict__ out_h1) {
  int i = blockIdx.x * 256 + threadIdx.x;
  if (i < BATCH * H_SZ) {
    out_h0[i] = hbuf[1 * BATCH * H_SZ + i];
    out_h1[i] = hbuf[2 * BATCH * H_SZ + i];
  }
}

extern "C" void kernel_launch(void* const* d_in, const int* in_sizes, int n_in,
                              void* d_out, int out_size, void* d_ws,
                              size_t ws_size, hipStream_t stream) {
  const int*   x     = (const int*)d_in[0];
  const float* emb   = (const float*)d_in[1];
  const float* W_ih0 = (const float*)d_in[2];
  const float* b_ih0 = (const float*)d_in[3];
  const float* W_hh0 = (const float*)d_in[4];
  const float* b_hh0 = (const float*)d_in[5];
  const float* W_ih1 = (const float*)d_in[6];
  const float* b_ih1 = (const float*)d_in[7];
  const float* W_hh1 = (const float*)d_in[8];
  const float* b_hh1 = (const float*)d_in[9];
  const float* W_out = (const float*)d_in[10];
  const float* b_out = (const float*)d_in[11];

  float* logits = (float*)d_out;
  float* out_h0 = logits + (size_t)M_ROWS * V_SZ;
  float* out_h1 = out_h0 + BATCH * H_SZ;

  float* X0   = (float*)d_ws;                       // 16 MB
  float* outs = X0 + (size_t)M_ROWS * H_SZ;         // 16 MB
  float* hbuf = outs + (size_t)M_ROWS * H_SZ;       // 128 KB
  unsigned* bar = (unsigned*)(hbuf + 4 * BATCH * H_SZ);

  (void)hipMemsetAsync(bar, 0, 2 * sizeof(unsigned), stream);

  dim3 g1(H_SZ / 128, M_ROWS / 64);                 // 8 x 64
  k_x0<<<g1, 256, 0, stream>>>(x, emb, W_ih0, b_ih0, b_hh0, X0);

  const size_t smem = (size_t)(3 * CPB + 2 * BATCH) * WROW * sizeof(float);
  k_recur<<<NBLK, 256, smem, stream>>>(X0, W_hh0, W_ih1, b_ih1, W_hh1, b_hh1,
                                       outs, hbuf, bar);

  dim3 g2(V_SZ / 128, M_ROWS / 64);                 // 250 x 64
  k_logits<<<g2, 256, 0, stream>>>(outs, W_out, b_out, logits);

  k_tail<<<(BATCH * H_SZ + 255) / 256, 256, 0, stream>>>(hbuf, out_h0, out_h1);
}